// LowpassDetector_42820823941642
// MI455X (gfx1250) — compile-verified
//
#include <hip/hip_runtime.h>
#include <hip/hip_bf16.h>

typedef float v8f __attribute__((ext_vector_type(8)));
typedef float v2f __attribute__((ext_vector_type(2)));

// ---------------------------------------------------------------------------
// Init kernel (1 thread): build block matrices from filter coefficients.
// ws float layout:
//   [0..255]   W0[m][k]  : block-0 matrix (y[0..4]=xf[0..4], recursion t>=5)
//   [256..511] W [m][k]  : steady-state impulse-response matrix
//   [512..575] S [m][j]  : response to carried-in state y[-4+j]
// a_rev (d_in[2]) = [A4, A3, A2, A1, -1]  =>  A_i = a_rev[4-i]
// ---------------------------------------------------------------------------
__global__ void lp_init_kernel(const float* __restrict__ arev,
                               float* __restrict__ ws) {
    if (threadIdx.x != 0 || blockIdx.x != 0) return;
    const float A1 = arev[3], A2 = arev[2], A3 = arev[1], A4 = arev[0];

    // W: standard recursion, zero initial state, unit impulse at k
    for (int k = 0; k < 16; ++k) {
        float y[16];
        for (int t = 0; t < 16; ++t) {
            float acc = (t == k) ? 1.0f : 0.0f;
            if (t >= 1) acc -= A1 * y[t - 1];
            if (t >= 2) acc -= A2 * y[t - 2];
            if (t >= 3) acc -= A3 * y[t - 3];
            if (t >= 4) acc -= A4 * y[t - 4];
            y[t] = acc;
        }
        for (int m = 0; m < 16; ++m) ws[256 + m * 16 + k] = y[m];
    }
    // W0: reference copies y[0..4] = xf[0..4]; recursion only for t >= 5
    for (int k = 0; k < 16; ++k) {
        float y[16];
        for (int t = 0; t < 16; ++t) {
            float acc = (t == k) ? 1.0f : 0.0f;
            if (t > 4) {
                acc -= A1 * y[t - 1];
                acc -= A2 * y[t - 2];
                acc -= A3 * y[t - 3];
                acc -= A4 * y[t - 4];
            }
            y[t] = acc;
        }
        for (int m = 0; m < 16; ++m) ws[0 + m * 16 + k] = y[m];
    }
    // S: homogeneous response to unit state y[-4+j]
    for (int j = 0; j < 4; ++j) {
        float yb[20];
        for (int i = 0; i < 20; ++i) yb[i] = 0.0f;
        yb[j] = 1.0f;
        for (int t = 0; t < 16; ++t)
            yb[4 + t] = -(A1 * yb[3 + t] + A2 * yb[2 + t] +
                          A3 * yb[1 + t] + A4 * yb[0 + t]);
        for (int m = 0; m < 16; ++m) ws[512 + m * 4 + j] = yb[4 + m];
    }
}

// ---------------------------------------------------------------------------
// Main kernel: one wave (32 threads) per 16-column tile; iterate T/16 blocks.
// Y_blk = W @ XF_blk + S @ state   via 5x V_WMMA_F32_16X16X4_F32 per block.
// ---------------------------------------------------------------------------
__global__ __launch_bounds__(32)
void lp_wmma_kernel(const float* __restrict__ sig,
                    const float* __restrict__ bco,
                    const float* __restrict__ ws,
                    float* __restrict__ out,
                    int T, int Bd) {
    const int lane = threadIdx.x;       // 0..31
    const int h    = lane >> 4;         // lane half (0/1)
    const int cl   = lane & 15;         // column within tile / matrix row M
    const int col  = blockIdx.x * 16 + cl;

    const float* x0 = sig;                         // signal[0]
    const float* x1 = sig + (size_t)T * Bd;        // signal[1]

    // FIR taps: xf[t] = b4*x[t] + b3*x[t-1] + b2*x[t-2] + b1*x[t-3] + b0*x[t-4]
    const float b0 = bco[0], b1 = bco[1], b2 = bco[2], b3 = bco[3], b4 = bco[4];

    // A-operands (16x4 f32): lane M = cl, VGPR0 holds K=2h, VGPR1 holds K=2h+1.
    // Start with W0 for block 0; reload W after block 0.
    v2f wA[4], sA;
#pragma unroll
    for (int g = 0; g < 4; ++g) {
        wA[g].x = ws[cl * 16 + 4 * g + 2 * h];
        wA[g].y = ws[cl * 16 + 4 * g + 2 * h + 1];
    }
    sA.x = ws[512 + cl * 4 + 2 * h];
    sA.y = ws[512 + cl * 4 + 2 * h + 1];

    // Carried state as a B-operand (4x16): stB.x = y[t0-4+2h], stB.y = y[t0-3+2h]
    v2f stB; stB.x = 0.0f; stB.y = 0.0f;

    // FIR history of sq (zero padding at t<0 matches reference)
    float s1 = 0.0f, s2 = 0.0f, s3 = 0.0f, s4 = 0.0f;

    const float* p0 = x0 + col;
    const float* p1 = x1 + col;
    float*       po = out + col + (size_t)(8 * h) * Bd;

    const int nblk = T >> 4;
    for (int blk = 0; blk < nblk; ++blk) {
        // ---- load the 16 new rows of both signals (coalesced 64B per row) ----
        float xa[16], xb[16];
#pragma unroll
        for (int r = 0; r < 16; ++r) {
            xa[r] = p0[(size_t)r * Bd];
            xb[r] = p1[(size_t)r * Bd];
        }

        // ---- square + 5-tap FIR with register-rolling history ----
        float xfv[16];
#pragma unroll
        for (int r = 0; r < 16; ++r) {
            float sq = fmaf(xa[r], xa[r], xb[r] * xb[r]);
            float xf = b4 * sq;
            xf = fmaf(b3, s1, xf);
            xf = fmaf(b2, s2, xf);
            xf = fmaf(b1, s3, xf);
            xf = fmaf(b0, s4, xf);
            xfv[r] = xf;
            s4 = s3; s3 = s2; s2 = s1; s1 = sq;
        }

        // ---- pack xf into B-operands (4x16 each): rows 4g+2h, 4g+2h+1 ----
        v2f bB[4];
#pragma unroll
        for (int g = 0; g < 4; ++g) {
            bB[g].x = h ? xfv[4 * g + 2] : xfv[4 * g + 0];
            bB[g].y = h ? xfv[4 * g + 3] : xfv[4 * g + 1];
        }

        // ---- Y = W @ XF + S @ state : 5 chained f32 WMMAs ----
        v8f acc = {0.f, 0.f, 0.f, 0.f, 0.f, 0.f, 0.f, 0.f};
        acc = __builtin_amdgcn_wmma_f32_16x16x4_f32(false, wA[0], false, bB[0],
                                                    (short)0, acc, false, false);
        acc = __builtin_amdgcn_wmma_f32_16x16x4_f32(false, wA[1], false, bB[1],
                                                    (short)0, acc, false, false);
        acc = __builtin_amdgcn_wmma_f32_16x16x4_f32(false, wA[2], false, bB[2],
                                                    (short)0, acc, false, false);
        acc = __builtin_amdgcn_wmma_f32_16x16x4_f32(false, wA[3], false, bB[3],
                                                    (short)0, acc, false, false);
        acc = __builtin_amdgcn_wmma_f32_16x16x4_f32(false, sA,    false, stB,
                                                    (short)0, acc, false, false);

        // ---- next state = Y rows 12..15 (D VGPRs 4..7, high lane half) ----
        {
            float a0 = __shfl(acc[4], cl + 16, 32);  // y[12] to low half
            float a1 = __shfl(acc[5], cl + 16, 32);  // y[13] to low half
            stB.x = h ? acc[6] : a0;                 // h=1 lanes own y[14]
            stB.y = h ? acc[7] : a1;                 // h=1 lanes own y[15]
        }

        // ---- store Y block: VGPR v holds rows v (lo half) / v+8 (hi half) ----
#pragma unroll
        for (int v = 0; v < 8; ++v) {
            po[(size_t)v * Bd] = acc[v];
        }

        // after block 0, switch W0 -> W
        if (blk == 0) {
#pragma unroll
            for (int g = 0; g < 4; ++g) {
                wA[g].x = ws[256 + cl * 16 + 4 * g + 2 * h];
                wA[g].y = ws[256 + cl * 16 + 4 * g + 2 * h + 1];
            }
        }

        p0 += (size_t)16 * Bd;
        p1 += (size_t)16 * Bd;
        po += (size_t)16 * Bd;
    }
}

extern "C" void kernel_launch(void* const* d_in, const int* in_sizes, int n_in,
                              void* d_out, int out_size, void* d_ws, size_t ws_size,
                              hipStream_t stream) {
    const float* sig  = (const float*)d_in[0];   // (2, T, B) f32
    const float* bco  = (const float*)d_in[1];   // (5,) f32
    const float* arev = (const float*)d_in[2];   // (5,) f32 = [A4,A3,A2,A1,-1]
    float*       out  = (float*)d_out;
    float*       ws   = (float*)d_ws;            // needs 576 floats

    const int Bd = 2048;
    const int T  = (int)((long)in_sizes[0] / (2L * Bd));   // 16384

    lp_init_kernel<<<1, 1, 0, stream>>>(arev, ws);
    lp_wmma_kernel<<<Bd / 16, 32, 0, stream>>>(sig, bco, ws, out, T, Bd);
}